// MSEObserver_8392366096875
// MI455X (gfx1250) — compile-verified
//
#include <hip/hip_runtime.h>

// MSEObserver 4-bit asymmetric per-tensor grid search, MI455X (gfx1250, wave32).
// K = 128 candidates (8 range steps x 16 zero-points). VALU-bound fused scoring;
// per-wave candidate reduction done with V_WMMA_F32_16X16X4_F32 (exact f32 rowsum
// against an all-ones B matrix).

#define GXB 256   // blocks along data dimension in scoring kernel
#define NTH 256   // threads per block (8 wave32)

typedef __attribute__((ext_vector_type(2))) float v2f;
typedef __attribute__((ext_vector_type(8))) float v8f;

// Monotone float <-> uint encoding (finite values) for integer atomic min/max.
__device__ __forceinline__ unsigned enc_f(float f) {
  unsigned u = __float_as_uint(f);
  return (u & 0x80000000u) ? ~u : (u | 0x80000000u);
}
__device__ __forceinline__ float dec_f(unsigned e) {
  unsigned u = (e & 0x80000000u) ? (e ^ 0x80000000u) : ~e;
  return __uint_as_float(u);
}

// Candidate k (0..127): i = k/16 + 1 (range step), zpc = k%16 (zero-point cand).
__device__ __forceinline__ void cand_params(int k, float xmin, float xmax,
                                            float& inv_s, float& zf, float& scale,
                                            float& nmin, float& nmax) {
  float xr     = xmax - xmin;
  float fi     = (float)((k >> 4) + 1);
  float tmax   = xr / 8.0f * fi;          // xrange / NUM * i
  float tdelta = tmax / 15.0f;            // / (QMAX - QMIN)
  float shift  = (float)(k & 15) * tdelta;
  nmin = fmaxf(0.0f - shift, xmin);
  nmax = fminf(tmax - shift, xmax);
  float min_neg = fminf(nmin, 0.0f);
  float max_pos = fmaxf(nmax, 0.0f);
  scale = fmaxf((max_pos - min_neg) / 15.0f, 1e-8f);
  float zp = 0.0f - rintf(min_neg / scale);          // round-half-even, as jnp.round
  zf = fminf(fmaxf(zp, 0.0f), 15.0f);
  inv_s = 1.0f / scale;
}

__global__ void k_init(unsigned* __restrict__ W) {
  if (threadIdx.x == 0) { W[0] = 0xFFFFFFFFu; W[1] = 0u; }
}

__global__ void k_minmax(const float* __restrict__ x, int n, unsigned* __restrict__ W) {
  float lmin = __int_as_float(0x7F800000);   // +inf
  float lmax = -lmin;
  int gid = blockIdx.x * NTH + threadIdx.x;
  int nt  = gridDim.x * NTH;
  int n4  = n >> 2;
  const float4* x4 = (const float4*)x;
  for (int i = gid; i < n4; i += nt) {
    float4 v = x4[i];
    lmin = fminf(lmin, fminf(fminf(v.x, v.y), fminf(v.z, v.w)));
    lmax = fmaxf(lmax, fmaxf(fmaxf(v.x, v.y), fmaxf(v.z, v.w)));
  }
  for (int i = (n4 << 2) + gid; i < n; i += nt) {
    float v = x[i]; lmin = fminf(lmin, v); lmax = fmaxf(lmax, v);
  }
#pragma unroll
  for (int m = 16; m >= 1; m >>= 1) {
    lmin = fminf(lmin, __shfl_xor(lmin, m, 32));
    lmax = fmaxf(lmax, __shfl_xor(lmax, m, 32));
  }
  __shared__ float smin[8], smax[8];
  int wid = threadIdx.x >> 5, lane = threadIdx.x & 31;
  if (lane == 0) { smin[wid] = lmin; smax[wid] = lmax; }
  __syncthreads();
  if (threadIdx.x == 0) {
    float bmin = smin[0], bmax = smax[0];
    for (int i = 1; i < 8; ++i) { bmin = fminf(bmin, smin[i]); bmax = fmaxf(bmax, smax[i]); }
    atomicMin(&W[0], enc_f(bmin));
    atomicMax(&W[1], enc_f(bmax));
  }
}

// Scoring: blockIdx.y selects a 32-candidate group; blockIdx.x tiles the data.
// Per-thread: acc[32] in VGPRs; per chunk of 8 elems, 32x8x5 VALU ops.
// Wave reduction: stage 32x32 acc values to LDS (stride 36 words, conflict-free),
// then 2 groups x 8 chained V_WMMA_F32_16X16X4_F32 with B = ones -> exact rowsums.
__global__ void __launch_bounds__(NTH) k_score(const float* __restrict__ x, int n,
                                               const unsigned* __restrict__ W,
                                               float* __restrict__ P) {
  __shared__ float cinv_s[32], czf_s[32];
  __shared__ float stage[8][32 * 36];
  __shared__ float wsum[8][32];

  const int tid = threadIdx.x;
  const int g   = blockIdx.y;

  if (tid < 32) {
    float inv_s, zf, sc, nm, nx;
    cand_params(g * 32 + tid, dec_f(W[0]), dec_f(W[1]), inv_s, zf, sc, nm, nx);
    cinv_s[tid] = inv_s; czf_s[tid] = zf;
  }
  __syncthreads();

  float cinv[32], czf[32], acc[32];
#pragma unroll
  for (int c = 0; c < 32; ++c) { cinv[c] = cinv_s[c]; czf[c] = czf_s[c]; acc[c] = 0.0f; }

  const int gid    = blockIdx.x * NTH + tid;
  const int stride = GXB * NTH * 8;
  for (int base = gid * 8; base + 8 <= n; base += stride) {
    const float4* p4 = (const float4*)(x + base);
    float4 v0 = p4[0], v1 = p4[1];
    float xv[8] = {v0.x, v0.y, v0.z, v0.w, v1.x, v1.y, v1.z, v1.w};
#pragma unroll
    for (int c = 0; c < 32; ++c) {
      const float is = cinv[c], z = czf[c];
#pragma unroll
      for (int e = 0; e < 8; ++e) {
        float w = fmaf(xv[e], is, z);                           // x/s + zp
        float q = __builtin_amdgcn_fmed3f(rintf(w), 0.0f, 15.0f); // clip(round(.))
        float d = q - w;                                        // (q-zp) - x/s
        acc[c] = fmaf(d, d, acc[c]);
      }
    }
  }
  // generic scalar tail (no-op when n % 8 == 0)
  for (int i = (n / 8) * 8 + gid; i < n; i += GXB * NTH) {
    float xe = x[i];
#pragma unroll
    for (int c = 0; c < 32; ++c) {
      float w = fmaf(xe, cinv[c], czf[c]);
      float q = __builtin_amdgcn_fmed3f(rintf(w), 0.0f, 15.0f);
      float d = q - w;
      acc[c] = fmaf(d, d, acc[c]);
    }
  }

  const int wid = tid >> 5, lane = tid & 31;
  float* sp = &stage[wid][0];
#pragma unroll
  for (int c = 0; c < 32; ++c) sp[c * 36 + lane] = acc[c];   // conflict-free stores

  // WMMA reduction: A = 16x4 f32 slices of stage (A layout: lane m -> K0/K1 in
  // vgpr0/1, lane m+16 -> K2/K3), B = ones (layout-independent), chained C-accum.
  const int lw = lane & 15, hh = lane >> 4;
  const v2f bones = {1.0f, 1.0f};
#pragma unroll
  for (int grp = 0; grp < 2; ++grp) {
    const int c0 = grp * 16;
    v8f dacc = {};
#pragma unroll
    for (int s = 0; s < 8; ++s) {
      const v2f* ap = (const v2f*)(sp + ((c0 + lw) * 36 + 4 * s + 2 * hh)); // 8B aligned
      v2f a = *ap;
      dacc = __builtin_amdgcn_wmma_f32_16x16x4_f32(false, a, false, bones,
                                                   (short)0, dacc, false, false);
    }
    // D layout: vgpr r, lanes 0-15 -> row r; lanes 16-31 -> row r+8 (all cols equal)
    if (lane == 0) {
#pragma unroll
      for (int r = 0; r < 8; ++r) wsum[wid][c0 + r] = dacc[r];
    } else if (lane == 16) {
#pragma unroll
      for (int r = 0; r < 8; ++r) wsum[wid][c0 + 8 + r] = dacc[r];
    }
  }
  __syncthreads();

  if (tid < 32) {
    float p = 0.0f;
#pragma unroll
    for (int wv = 0; wv < 8; ++wv) p += wsum[wv][tid];        // fixed order: deterministic
    P[(g * GXB + blockIdx.x) * 32 + tid] = p;
  }
}

__global__ void k_final(const unsigned* __restrict__ W, const float* __restrict__ P,
                        float* __restrict__ out) {
  const int k = threadIdx.x;           // 128 threads, one per candidate
  const float xmin = dec_f(W[0]), xmax = dec_f(W[1]);
  float inv_s, zf, sc, nm, nx;
  cand_params(k, xmin, xmax, inv_s, zf, sc, nm, nx);
  const int g = k >> 5, c = k & 31;
  float s = 0.0f;
  for (int bx = 0; bx < GXB; ++bx) s += P[(g * GXB + bx) * 32 + c];  // fixed order
  const float score = sc * sc * s;     // mean's 1/N factor is monotone -> dropped

  __shared__ float sscore[128], snm[128], snx[128];
  sscore[k] = score; snm[k] = nm; snx[k] = nx;
  __syncthreads();
  if (k == 0) {
    int best = 0; float bs = sscore[0];
    for (int i = 1; i < 128; ++i)
      if (sscore[i] < bs) { bs = sscore[i]; best = i; }       // strict '<': first occurrence
    out[0] = snm[best];
    out[1] = snx[best];
  }
}

extern "C" void kernel_launch(void* const* d_in, const int* in_sizes, int n_in,
                              void* d_out, int out_size, void* d_ws, size_t ws_size,
                              hipStream_t stream) {
  (void)n_in; (void)out_size; (void)ws_size;
  const float* x = (const float*)d_in[0];
  const int n = in_sizes[0];

  unsigned* W = (unsigned*)d_ws;                 // W[0]=min_enc, W[1]=max_enc
  float* P = (float*)d_ws + 16;                  // 4*GXB*32 floats of block partials
  float* out = (float*)d_out;

  k_init<<<1, 64, 0, stream>>>(W);
  k_minmax<<<1024, NTH, 0, stream>>>(x, n, W);
  dim3 gs(GXB, 4, 1);
  k_score<<<gs, NTH, 0, stream>>>(x, n, W, P);
  k_final<<<1, 128, 0, stream>>>(W, P, out);
}